// IterNormSigmaSingle_51127290691774
// MI455X (gfx1250) — compile-verified
//
#include <hip/hip_runtime.h>
#include <hip/hip_bf16.h>
#include <math.h>

// IterNorm (ZCA whitening, Newton-Schulz) for x[64,64,112,112] fp32.
// Pass 1: G = X X^T and per-channel sums (WMMA f32 16x16x4, async-LDS double buffer).
// Pass 2: tiny 64x64 Newton-Schulz on one workgroup -> wm, bias = wm@mean.
// Pass 3: out = wm @ x - bias (WMMA f32 16x16x4, async-LDS double buffer).

#define CCH   64
#define HW    12544            // 112*112
#define MCOLS 802816           // 64*HW
#define NCHUNK 12544           // MCOLS / 64
#define LP    68               // LDS pitch in floats (conflict-free, 16B-aligned)

typedef __attribute__((ext_vector_type(2))) float v2f;
typedef __attribute__((ext_vector_type(8))) float v8f;

// ---- CDNA5 async global->LDS copy (ASYNCcnt path), bypasses VGPRs ----
__device__ __forceinline__ void async_copy_b128(const float* gsrc, float* ldst) {
    // Low 32 bits of a generic pointer to __shared__ are the LDS byte address
    unsigned loff = (unsigned)(size_t)ldst;
    asm volatile("global_load_async_to_lds_b128 %0, %1, off"
                 :: "v"(loff), "v"(gsrc)
                 : "memory");
}
__device__ __forceinline__ void wait_async0() {
    asm volatile("s_wait_asynccnt 0x0" ::: "memory");
}

// issue async staging of one 64-channel x 64-col chunk into an LDS tile
__device__ __forceinline__ void stage_chunk_async(const float* __restrict__ x,
                                                  int chunk, float* dst,
                                                  int r0, int c4) {
    const int J    = chunk * 64;          // never straddles a batch (HW%64==0)
    const int b    = J / HW;
    const int off0 = J % HW;
    const float* xb = x + (size_t)b * CCH * HW + off0;
    #pragma unroll
    for (int pass = 0; pass < 4; ++pass) {
        const int row = pass * 16 + r0;
        async_copy_b128(xb + (size_t)row * HW + c4, dst + row * LP + c4);
    }
}

__global__ void IterNorm_zero_kernel(float* __restrict__ p, int n) {
    int i = blockIdx.x * blockDim.x + threadIdx.x;
    if (i < n) p[i] = 0.0f;
}

// ---------------- Pass 1: Gram matrix + channel sums ----------------
__global__ __launch_bounds__(256) void IterNorm_gram_kernel(
    const float* __restrict__ x, float* __restrict__ G, float* __restrict__ sums)
{
    __shared__ float ls[2][CCH * LP];
    const int t    = threadIdx.x;
    const int lane = t & 31;
    const int wave = t >> 5;          // 0..7
    const int hl   = lane >> 4;       // half-wave (0/1)
    const int l15  = lane & 15;

    // each wave owns two 16x16 tiles of the 4x4 tile grid of G
    const int ta  = 2 * wave, tb = 2 * wave + 1;
    const int ci0 = ta >> 2, cj0 = ta & 3;
    const int ci1 = tb >> 2, cj1 = tb & 3;

    v8f acc0 = {}; v8f acc1 = {};

    // channel-sum partial: thread t sums channel (t&63), column quarter (t>>6)
    float sloc = 0.0f;
    const int sc = t & 63;
    const int sq = (t >> 6) * 16;

    const int r0 = t >> 4;            // staging row within pass
    const int c4 = (t & 15) * 4;      // staging col (float4)

    int chunk = blockIdx.x;
    if (chunk < NCHUNK) stage_chunk_async(x, chunk, ls[0], r0, c4);

    int buf = 0;
    for (; chunk < NCHUNK; chunk += gridDim.x) {
        wait_async0();
        __syncthreads();                       // ls[buf] ready; ls[buf^1] free

        const int nxt = chunk + gridDim.x;
        if (nxt < NCHUNK) stage_chunk_async(x, nxt, ls[buf ^ 1], r0, c4);

        const float* lb = ls[buf];

        // channel partial sums
        #pragma unroll
        for (int jj = 0; jj < 16; ++jj) sloc += lb[sc * LP + sq + jj];

        // accumulate Gram tiles over 16 K-chunks of 4 columns
        #pragma unroll
        for (int jc = 0; jc < 16; ++jc) {
            const int jb = jc * 4 + hl * 2;   // lanes<16: K{0,1}, lanes>=16: K{2,3}
            v2f A0, B0, A1, B1;
            { const float* p_ = &lb[(ci0*16 + l15) * LP + jb]; A0.x = p_[0]; A0.y = p_[1]; }
            { const float* p_ = &lb[(cj0*16 + l15) * LP + jb]; B0.x = p_[0]; B0.y = p_[1]; }
            { const float* p_ = &lb[(ci1*16 + l15) * LP + jb]; A1.x = p_[0]; A1.y = p_[1]; }
            { const float* p_ = &lb[(cj1*16 + l15) * LP + jb]; B1.x = p_[0]; B1.y = p_[1]; }
            acc0 = __builtin_amdgcn_wmma_f32_16x16x4_f32(false, A0, false, B0,
                                                         (short)0, acc0, false, false);
            acc1 = __builtin_amdgcn_wmma_f32_16x16x4_f32(false, A1, false, B1,
                                                         (short)0, acc1, false, false);
        }
        __syncthreads();                       // done reading ls[buf]
        buf ^= 1;
    }

    // C/D layout: VGPR r -> M = r (+8 for lanes 16..31), N = lane%16
    #pragma unroll
    for (int r = 0; r < 8; ++r) {
        const int mm = r + hl * 8;
        const int nn = l15;
        atomicAdd(&G[(ci0*16 + mm) * CCH + cj0*16 + nn], acc0[r]);
        atomicAdd(&G[(ci1*16 + mm) * CCH + cj1*16 + nn], acc1[r]);
    }
    atomicAdd(&sums[sc], sloc);
}

// ---------------- Pass 2: sigma -> Newton-Schulz -> wm, bias ----------------
__global__ __launch_bounds__(256) void IterNorm_newton_kernel(
    const float* __restrict__ G, const float* __restrict__ sums,
    float* __restrict__ wm, float* __restrict__ bias, float* __restrict__ tmp)
{
    __shared__ float sig[4096];
    __shared__ float p[4096];
    __shared__ float tA[4096];
    __shared__ float s[64];
    __shared__ float red[64];
    __shared__ float rsh;
    const int t = threadIdx.x;
    const float inv_m = 1.0f / (float)MCOLS;

    if (t < 64) s[t] = sums[t];
    __syncthreads();

    for (int e = t; e < 4096; e += 256) {
        const int i = e >> 6, j = e & 63;
        float v = (G[e] - s[i] * s[j] * inv_m) * inv_m;
        if (i == j) v += 1e-5f;
        sig[e] = v;
    }
    __syncthreads();
    if (t < 64) red[t] = sig[t * 64 + t];
    __syncthreads();
    if (t == 0) {
        float tr = 0.0f;
        for (int i = 0; i < 64; ++i) tr += red[i];
        rsh = 1.0f / tr;
    }
    __syncthreads();
    const float r = rsh;
    for (int e = t; e < 4096; e += 256) {
        sig[e] *= r;
        p[e] = ((e >> 6) == (e & 63)) ? 1.0f : 0.0f;
    }
    __syncthreads();

    for (int iter = 0; iter < 5; ++iter) {
        for (int e = t; e < 4096; e += 256) {       // tA = p @ p
            const int i = e >> 6, j = e & 63;
            float a = 0.0f;
            for (int k = 0; k < 64; ++k) a += p[i*64+k] * p[k*64+j];
            tA[e] = a;
        }
        __syncthreads();
        for (int e = t; e < 4096; e += 256) {       // tmp = tA @ p  (= p^3)
            const int i = e >> 6, j = e & 63;
            float a = 0.0f;
            for (int k = 0; k < 64; ++k) a += tA[i*64+k] * p[k*64+j];
            tmp[e] = a;
        }
        __syncthreads();
        for (int e = t; e < 4096; e += 256) {       // tA = p^3 @ sigma_n
            const int i = e >> 6, j = e & 63;
            float a = 0.0f;
            for (int k = 0; k < 64; ++k) a += tmp[i*64+k] * sig[k*64+j];
            tA[e] = a;
        }
        __syncthreads();
        for (int e = t; e < 4096; e += 256)
            p[e] = 1.5f * p[e] - 0.5f * tA[e];
        __syncthreads();
    }

    const float sr = sqrtf(r);
    for (int e = t; e < 4096; e += 256) wm[e] = p[e] * sr;
    __syncthreads();
    if (t < 64) {                                    // bias = wm @ mean
        float a = 0.0f;
        for (int k = 0; k < 64; ++k) a += p[t*64+k] * sr * (s[k] * inv_m);
        bias[t] = a;
    }
}

// ---------------- Pass 3: out = wm @ x - bias ----------------
__global__ __launch_bounds__(256) void IterNorm_apply_kernel(
    const float* __restrict__ x, const float* __restrict__ wm,
    const float* __restrict__ bias, float* __restrict__ out)
{
    __shared__ float ls[2][CCH * LP];  // x tiles (double buffered)
    __shared__ float wls[CCH * LP];    // wm, pitched
    __shared__ float bls[64];
    const int t    = threadIdx.x;
    const int lane = t & 31;
    const int wave = t >> 5;
    const int hl   = lane >> 4;
    const int l15  = lane & 15;

    for (int e = t; e < 4096; e += 256)
        wls[(e >> 6) * LP + (e & 63)] = wm[e];
    if (t < 64) bls[t] = bias[t];

    // each wave owns two output tiles: tile id -> (col-tile jt, channel-tile ci)
    const int ta  = 2 * wave, tb = 2 * wave + 1;
    const int jt0 = ta >> 2, ci0 = ta & 3;
    const int jt1 = tb >> 2, ci1 = tb & 3;

    const int r0 = t >> 4;
    const int c4 = (t & 15) * 4;

    int chunk = blockIdx.x;
    if (chunk < NCHUNK) stage_chunk_async(x, chunk, ls[0], r0, c4);

    int buf = 0;
    for (; chunk < NCHUNK; chunk += gridDim.x) {
        wait_async0();
        __syncthreads();                       // ls[buf] (and wls on iter 0) ready

        const int nxt = chunk + gridDim.x;
        if (nxt < NCHUNK) stage_chunk_async(x, nxt, ls[buf ^ 1], r0, c4);

        const float* lb = ls[buf];
        const int J    = chunk * 64;
        const int b    = J / HW;
        const int off0 = J % HW;
        float* ob = out + (size_t)b * CCH * HW + off0;

        v8f acc0 = {}; v8f acc1 = {};
        #pragma unroll
        for (int kc = 0; kc < 16; ++kc) {
            const int kb = kc * 4 + hl * 2;   // A: K{0,1}/{2,3}; B rows K=kb, kb+1
            v2f A0, A1, B0, B1;
            { const float* p_ = &wls[(ci0*16 + l15) * LP + kb]; A0.x = p_[0]; A0.y = p_[1]; }
            { const float* p_ = &wls[(ci1*16 + l15) * LP + kb]; A1.x = p_[0]; A1.y = p_[1]; }
            { const float* p_ = &lb[kb * LP + jt0*16 + l15];    B0.x = p_[0]; B0.y = p_[LP]; }
            { const float* p_ = &lb[kb * LP + jt1*16 + l15];    B1.x = p_[0]; B1.y = p_[LP]; }
            acc0 = __builtin_amdgcn_wmma_f32_16x16x4_f32(false, A0, false, B0,
                                                         (short)0, acc0, false, false);
            acc1 = __builtin_amdgcn_wmma_f32_16x16x4_f32(false, A1, false, B1,
                                                         (short)0, acc1, false, false);
        }

        // C/D layout: VGPR r -> channel row r (+8 for upper half-wave)
        #pragma unroll
        for (int r = 0; r < 8; ++r) {
            { const int c = ci0*16 + r + hl*8;
              ob[(size_t)c * HW + jt0*16 + l15] = acc0[r] - bls[c]; }
            { const int c = ci1*16 + r + hl*8;
              ob[(size_t)c * HW + jt1*16 + l15] = acc1[r] - bls[c]; }
        }
        __syncthreads();                       // done reading ls[buf]
        buf ^= 1;
    }
}

extern "C" void kernel_launch(void* const* d_in, const int* in_sizes, int n_in,
                              void* d_out, int out_size, void* d_ws, size_t ws_size,
                              hipStream_t stream) {
    const float* x  = (const float*)d_in[0];
    float* out = (float*)d_out;
    float* ws  = (float*)d_ws;

    float* G    = ws;              // 4096
    float* sums = ws + 4096;       // 64
    float* wm   = ws + 4224;       // 4096
    float* bias = ws + 8320;       // 64
    float* tmp  = ws + 8448;       // 4096

    IterNorm_zero_kernel<<<(4160 + 255) / 256, 256, 0, stream>>>(ws, 4160);
    IterNorm_gram_kernel<<<1024, 256, 0, stream>>>(x, G, sums);
    IterNorm_newton_kernel<<<1, 256, 0, stream>>>(G, sums, wm, bias, tmp);
    IterNorm_apply_kernel<<<2048, 256, 0, stream>>>(x, wm, bias, out);
}